// AdapFilter3d_50483045597348
// MI455X (gfx1250) — compile-verified
//
#include <hip/hip_runtime.h>

// Problem constants (fixed by the reference's setup_inputs)
#define B_  2
#define C_  16
#define D_  32
#define H_  64
#define W_  64
#define TY  16            // output rows per block
#define TZT 3             // tile z extent (z-1..z+1)
#define TYT (TY + 2)      // tile y extent (halo)
#define TXT 66            // tile x extent (halo)
#define TXS 68            // padded LDS row stride (floats)

typedef float v4f __attribute__((ext_vector_type(4)));
typedef __attribute__((address_space(1))) int  g_i32;   // global int
typedef __attribute__((address_space(3))) int  l_i32;   // LDS int
typedef __attribute__((address_space(3))) float lds_f;

// ---- CDNA5 async global -> LDS copy (ASYNCcnt-tracked) -------------------
__device__ __forceinline__ void async_g2lds_b32(const float* gsrc, float* lds_dst) {
#if __has_builtin(__builtin_amdgcn_global_load_async_to_lds_b32)
    __builtin_amdgcn_global_load_async_to_lds_b32(
        (g_i32*)gsrc, (l_i32*)lds_dst, /*offset=*/0, /*cpol=*/0);
#else
    unsigned ldsaddr = (unsigned)(size_t)(lds_f*)lds_dst;
    asm volatile("global_load_async_to_lds_b32 %0, %1, off"
                 :: "v"(ldsaddr), "v"((unsigned long long)(size_t)gsrc)
                 : "memory");
#endif
}

__device__ __forceinline__ void wait_async0() {
#if __has_builtin(__builtin_amdgcn_s_wait_asynccnt)
    __builtin_amdgcn_s_wait_asynccnt(0);
#else
    asm volatile("s_wait_asynccnt 0" ::: "memory");
#endif
}

// --------------------------------------------------------------------------
__global__ __launch_bounds__(256) void AdapFilter3d_50483045597348_kernel(
    const float* __restrict__ x, const float* __restrict__ F,
    float* __restrict__ out)
{
    __shared__ float tile[TZT * TYT * TXS];   // 3*18*68*4 = 14688 B

    const int tid = threadIdx.x;
    const int y0  = blockIdx.x * TY;          // first output row of this block
    const int z   = blockIdx.y;
    const int bc  = blockIdx.z;               // flattened (b, c)

    // Per-thread output pixels: 4 consecutive x
    const int tx = tid & 15;
    const int ty = tid >> 4;
    const int x0 = tx * 4;
    const int y  = y0 + ty;
    const size_t p0 = (((size_t)bc * D_ + z) * H_ + y) * W_ + x0;
    const v4f* Fv = (const v4f*)(F + p0 * 27);   // 432 B, 16B-aligned (x0 % 4 == 0)

    // Warm the F stream (gfx1250 global_prefetch) while the LDS tile fills.
    __builtin_prefetch((const void*)Fv, 0, 0);
    __builtin_prefetch((const char*)Fv + 256, 0, 0);

    // ---- Stage padded input halo tile into LDS via async copies ----------
    const float* xbc = x + (size_t)bc * (D_ * H_ * W_);
    for (int t = tid; t < TZT * TYT * TXT; t += 256) {
        const int zz = t / (TYT * TXT);
        const int r  = t - zz * (TYT * TXT);
        const int yy = r / TXT;
        const int xx = r - yy * TXT;
        const int gz = z  - 1 + zz;
        const int gy = y0 - 1 + yy;
        const int gx = xx - 1;
        const int lidx = (zz * TYT + yy) * TXS + xx;
        if ((unsigned)gz < (unsigned)D_ &&
            (unsigned)gy < (unsigned)H_ &&
            (unsigned)gx < (unsigned)W_) {
            async_g2lds_b32(xbc + ((size_t)gz * H_ + gy) * W_ + gx, &tile[lidx]);
        } else {
            tile[lidx] = 0.0f;                // zero padding (ConstantPad3d)
        }
    }
    wait_async0();          // my async copies landed in LDS
    __syncthreads();        // everyone's copies + zero stores visible

    // ---- 27-tap adaptive filter for 4 pixels, F via NT b128 loads --------
    float acc[4] = {0.f, 0.f, 0.f, 0.f};
#pragma unroll
    for (int jv = 0; jv < 27; ++jv) {
        const v4f fv = __builtin_nontemporal_load(Fv + jv);
#pragma unroll
        for (int e = 0; e < 4; ++e) {
            const int f  = jv * 4 + e;          // flat index into 4*27 F block
            const int p  = f / 27;              // which of my 4 pixels (compile-time)
            const int k  = f - p * 27;          // tap id 0..26      (compile-time)
            const int ki = k / 9;               // z offset
            const int kj = (k - ki * 9) / 3;    // y offset
            const int kk = k - ki * 9 - kj * 3; // x offset
            const float xin =
                tile[(ki * TYT + (ty + kj)) * TXS + (x0 + p + kk)];
            acc[p] += fv[e] * xin;
        }
    }

    v4f ov = {acc[0], acc[1], acc[2], acc[3]};
    __builtin_nontemporal_store(ov, (v4f*)(out + p0));   // streamed store

}

// --------------------------------------------------------------------------
extern "C" void kernel_launch(void* const* d_in, const int* in_sizes, int n_in,
                              void* d_out, int out_size, void* d_ws, size_t ws_size,
                              hipStream_t stream) {
    (void)in_sizes; (void)n_in; (void)out_size; (void)d_ws; (void)ws_size;
    const float* x = (const float*)d_in[0];
    const float* F = (const float*)d_in[1];
    float* out     = (float*)d_out;

    dim3 grid(H_ / TY, D_, B_ * C_);   // (4, 32, 32) blocks
    dim3 block(256);                   // 8 wave32s
    AdapFilter3d_50483045597348_kernel<<<grid, block, 0, stream>>>(x, F, out);
}